// MultiLayer_54915451846965
// MI455X (gfx1250) — compile-verified
//
#include <hip/hip_runtime.h>
#include <cstdint>

#define NMAT     96
#define CHUNK    2                       // matrices staged per chunk
#define NCHUNK   (NMAT / CHUNK)          // 48 chunks
#define LANES    32
#define WPB      8                       // wave32s per block
#define THREADS  (WPB * LANES)           // 256
#define QUADS    4                       // 16B rows per 4x4 f32 matrix
#define F4_PER_BUF   (CHUNK * QUADS * LANES)   // 256 float4 = 4 KB per buffer
#define F4_PER_WAVE  (2 * F4_PER_BUF)          // double buffered, 8 KB per wave

typedef float v4f __attribute__((ext_vector_type(4)));

// ---- CDNA5 async global->LDS copy (ASYNCcnt-tracked), per-lane 16B ----
// Streamed-once data: non-temporal hint so 805 MB doesn't thrash WGP$/L2.
__device__ __forceinline__ void async_ld_b128_nt(unsigned lds_addr, const void* gaddr) {
    asm volatile("global_load_async_to_lds_b128 %0, %1, off th:TH_LOAD_NT"
                 :: "v"(lds_addr), "v"(gaddr)
                 : "memory");
}

#define S_WAIT_ASYNCCNT(N) asm volatile("s_wait_asynccnt " #N ::: "memory")

__device__ __forceinline__ void issue_chunk(unsigned lds_base, const char* gbase,
                                            int chunkIdx, int buf, int lane) {
#pragma unroll
    for (int s = 0; s < CHUNK; ++s) {
#pragma unroll
        for (int q = 0; q < QUADS; ++q) {
            unsigned ldst = lds_base +
                (unsigned)(buf * F4_PER_BUF + (s * QUADS + q) * LANES + lane) * 16u;
            const char* g = gbase + (size_t)(chunkIdx * CHUNK + s) * 64u + (size_t)q * 16u;
            async_ld_b128_nt(ldst, g);
        }
    }
}

// fold CHUNK staged matrices into the carry (64 FMAs per matrix per lane)
__device__ __forceinline__ void fold_chunk(const float4* __restrict__ b, int lane,
                                           float cr[4][4]) {
#pragma unroll
    for (int s = 0; s < CHUNK; ++s) {
        // rows of M; ds_load_b128 with 16B lane stride: bank-conflict-free
        float4 m0 = b[(s * QUADS + 0) * LANES + lane];
        float4 m1 = b[(s * QUADS + 1) * LANES + lane];
        float4 m2 = b[(s * QUADS + 2) * LANES + lane];
        float4 m3 = b[(s * QUADS + 3) * LANES + lane];
#pragma unroll
        for (int i = 0; i < 4; ++i) {
            float a0 = cr[i][0], a1 = cr[i][1], a2 = cr[i][2], a3 = cr[i][3];
            cr[i][0] = fmaf(a3, m3.x, fmaf(a2, m2.x, fmaf(a1, m1.x, a0 * m0.x)));
            cr[i][1] = fmaf(a3, m3.y, fmaf(a2, m2.y, fmaf(a1, m1.y, a0 * m0.y)));
            cr[i][2] = fmaf(a3, m3.z, fmaf(a2, m2.z, fmaf(a1, m1.z, a0 * m0.z)));
            cr[i][3] = fmaf(a3, m3.w, fmaf(a2, m2.w, fmaf(a1, m1.w, a0 * m0.w)));
        }
    }
}

__global__ __launch_bounds__(THREADS)
void MultiLayer_chain4x4_kernel(const float* __restrict__ mats,
                                float* __restrict__ out,
                                int nBatch) {
    __shared__ float4 smem[WPB * F4_PER_WAVE];   // 64 KB

    const int lane  = threadIdx.x & (LANES - 1);
    const int wave  = threadIdx.x >> 5;
    const int batch = blockIdx.x * THREADS + (int)threadIdx.x;
    if (batch >= nBatch) return;

    const char* gbase = (const char*)mats + (size_t)batch * (NMAT * 16 * sizeof(float));
    float4* wbuf = &smem[wave * F4_PER_WAVE];
    // low 32 bits of the flat address of a __shared__ object == LDS byte offset
    const unsigned lds_base = (unsigned)(uintptr_t)wbuf;

    // carry = identity; fold all 96 matrices (I*M0*...*M95 == reference chain)
    float cr[4][4];
#pragma unroll
    for (int i = 0; i < 4; ++i)
#pragma unroll
        for (int j = 0; j < 4; ++j)
            cr[i][j] = (i == j) ? 1.0f : 0.0f;

    // prologue: stage chunk 0 into buffer 0
    issue_chunk(lds_base, gbase, 0, 0, lane);

    // steady state: issue next chunk, wait for current (in-order retirement:
    // ASYNCcnt <= 8 means everything except the 8 just-issued loads is done)
    for (int ch = 0; ch < NCHUNK - 1; ++ch) {
        issue_chunk(lds_base, gbase, ch + 1, (ch + 1) & 1, lane);
        S_WAIT_ASYNCCNT(8);
        fold_chunk(wbuf + (ch & 1) * F4_PER_BUF, lane, cr);
    }

    // epilogue: drain and fold the last chunk
    S_WAIT_ASYNCCNT(0);
    fold_chunk(wbuf + ((NCHUNK - 1) & 1) * F4_PER_BUF, lane, cr);

    // non-temporal 64B contiguous store per lane
    v4f* o = (v4f*)((float4*)out + (size_t)batch * 4);
#pragma unroll
    for (int i = 0; i < 4; ++i) {
        v4f r = { cr[i][0], cr[i][1], cr[i][2], cr[i][3] };
        __builtin_nontemporal_store(r, o + i);
    }
}

extern "C" void kernel_launch(void* const* d_in, const int* in_sizes, int n_in,
                              void* d_out, int out_size, void* d_ws, size_t ws_size,
                              hipStream_t stream) {
    const float* mats = (const float*)d_in[0];
    float* out = (float*)d_out;
    const int nBatch = in_sizes[0] / (NMAT * 16);      // 131072
    const int blocks = (nBatch + THREADS - 1) / THREADS;
    MultiLayer_chain4x4_kernel<<<blocks, THREADS, 0, stream>>>(mats, out, nBatch);
}